// ScorePredictor_78262894068330
// MI455X (gfx1250) — compile-verified
//
#include <hip/hip_runtime.h>

typedef __attribute__((ext_vector_type(2))) float v2f;
typedef __attribute__((ext_vector_type(8))) float v8f;
typedef __attribute__((ext_vector_type(4))) float f4;

#define DIM 2048
#define KC 64
#define LDSTR 68   // 64 + 4 floats pad -> conflict-free striped b64 reads

__global__ __launch_bounds__(32)
void edge_score_wmma(const float* __restrict__ xsrc,
                     const float* __restrict__ xdst,
                     const float* __restrict__ rel,
                     const int*   __restrict__ srcIdx,
                     const int*   __restrict__ dstIdx,
                     float*       __restrict__ out,
                     int nEdges)
{
    __shared__ float ldsH[16 * LDSTR];
    __shared__ float ldsT[16 * LDSTR];
    __shared__ int   ldsRow[32];

    const int lane = threadIdx.x;      // 0..31, wave32
    const int e0   = blockIdx.x * 16;  // 16 edges per wave

    // stage 16 src row ids (lanes 0-15) and 16 dst row ids (lanes 16-31)
    {
        int e = e0 + (lane & 15);
        if (e >= nEdges) e = nEdges - 1;      // clamp for safety (E%16==0 normally)
        ldsRow[lane] = (lane < 16) ? srcIdx[e] : dstIdx[e];
    }
    __syncthreads();

    const int hi = lane >> 4;   // 0/1: selects K pair inside the 16x4 A tile
    const int m  = lane & 15;   // row (A) / col (B) index

    v8f acc = {};               // 16x16 f32 accumulator (diag = edge scores)

    for (int k0 = 0; k0 < DIM; k0 += KC) {
        // ---- stage: 16 head rows (pre-multiplied by rel) + 16 tail rows,
        // KC floats each. 8 iters x float4/lane, half-wave coalesced per row.
        #pragma unroll
        for (int j = 0; j < 8; ++j) {
            int row = 2 * j + hi;                  // 0..15
            int col = k0 + m * 4;                  // float4 column
            int rs  = ldsRow[row];
            int rt  = ldsRow[16 + row];
            f4 h = *(const f4*)(xsrc + (size_t)rs * DIM + col);
            f4 r = *(const f4*)(rel + col);
            f4 t = *(const f4*)(xdst + (size_t)rt * DIM + col);
            h.x *= r.x; h.y *= r.y; h.z *= r.z; h.w *= r.w;
            *(f4*)(&ldsH[row * LDSTR + m * 4]) = h;
            *(f4*)(&ldsT[row * LDSTR + m * 4]) = t;
        }
        __syncthreads();

        // ---- compute: KC/4 = 16 WMMA f32 16x16x4 steps (fp32-exact).
        // A operand: lane m holds A[m][kk+hi*2 .. +1]  (16x4 striped layout)
        // B operand: lane m holds B[kk+hi*2 .. +1][m] = tails[m][k...] (mirror)
        #pragma unroll
        for (int kk = 0; kk < KC; kk += 4) {
            int kl = kk + hi * 2;
            v2f a = *(const v2f*)(&ldsH[m * LDSTR + kl]);
            v2f b = *(const v2f*)(&ldsT[m * LDSTR + kl]);
            acc = __builtin_amdgcn_wmma_f32_16x16x4_f32(
                false, a, false, b, (short)0, acc, false, false);
        }
        __syncthreads();
    }

    // ---- extract diagonal: score_i is in VGPR (i&7), lane (i<8 ? i : i+16)
    float my = 0.0f;
    #pragma unroll
    for (int r = 0; r < 8; ++r) {
        float slo = __shfl(acc[r], r, 32);        // score_r
        float shi = __shfl(acc[r], r + 24, 32);   // score_{r+8}
        if (lane == r)     my = slo;
        if (lane == r + 8) my = shi;
    }
    if (lane < 16) {
        int e = e0 + lane;
        if (e < nEdges)
            out[e] = fminf(fmaxf(my, 0.0f), 1.0f);
    }
}

extern "C" void kernel_launch(void* const* d_in, const int* in_sizes, int n_in,
                              void* d_out, int out_size, void* d_ws, size_t ws_size,
                              hipStream_t stream) {
    const float* x_drug  = (const float*)d_in[0];
    const float* x_prot  = (const float*)d_in[1];
    const float* rel_ddi = (const float*)d_in[2];
    const float* rel_dpi = (const float*)d_in[3];
    const int* ddi_src = (const int*)d_in[4];
    const int* ddi_dst = (const int*)d_in[5];
    const int* dpi_src = (const int*)d_in[6];
    const int* dpi_dst = (const int*)d_in[7];
    const int* ppi_src = (const int*)d_in[8];
    const int* ppi_dst = (const int*)d_in[9];
    float* out = (float*)d_out;

    const int E0 = in_sizes[4];   // ddi edges
    const int E1 = in_sizes[6];   // dpi edges
    const int E2 = in_sizes[8];   // ppi edges

    dim3 blk(32);
    edge_score_wmma<<<(E0 + 15) / 16, blk, 0, stream>>>(
        x_drug, x_drug, rel_ddi, ddi_src, ddi_dst, out, E0);
    edge_score_wmma<<<(E1 + 15) / 16, blk, 0, stream>>>(
        x_drug, x_prot, rel_dpi, dpi_src, dpi_dst, out + E0, E1);
    // reference (faithfully) applies the DPI relation to PPI edges
    edge_score_wmma<<<(E2 + 15) / 16, blk, 0, stream>>>(
        x_prot, x_prot, rel_dpi, ppi_src, ppi_dst, out + E0 + E1, E2);
}